// TopKLogitAdjustedLoss_1864015807137
// MI455X (gfx1250) — compile-verified
//
#include <hip/hip_runtime.h>
#include <cstdint>

#define NUM_C 10000
#define NUM_C4 (NUM_C / 4)
#define NUM_B 4096
#define BLOCK 256
#define NWAVES (BLOCK / 32)

// Order-preserving float -> uint key (ascending key == ascending float)
__device__ __forceinline__ uint32_t f2key(float f) {
    uint32_t b = __float_as_uint(f);
    uint32_t m = (uint32_t)((int32_t)b >> 31);   // 0xFFFFFFFF if negative else 0
    return b ^ (m | 0x80000000u);
}

__device__ __forceinline__ float waveMax(float v) {
#pragma unroll
    for (int off = 16; off > 0; off >>= 1)
        v = fmaxf(v, __shfl_xor(v, off, 32));
    return v;
}
__device__ __forceinline__ float waveSum(float v) {
#pragma unroll
    for (int off = 16; off > 0; off >>= 1)
        v += __shfl_xor(v, off, 32);
    return v;
}

__device__ __forceinline__ float blockMax(float v, float* shred) {
    __syncthreads();
    v = waveMax(v);
    int wid = threadIdx.x >> 5, lane = threadIdx.x & 31;
    if (lane == 0) shred[wid] = v;
    __syncthreads();
    if (wid == 0) {
        float x = (lane < NWAVES) ? shred[lane] : -3.402823466e38f;
        x = waveMax(x);
        if (lane == 0) shred[0] = x;
    }
    __syncthreads();
    return shred[0];
}
__device__ __forceinline__ float blockSum(float v, float* shred) {
    __syncthreads();
    v = waveSum(v);
    int wid = threadIdx.x >> 5, lane = threadIdx.x & 31;
    if (lane == 0) shred[wid] = v;
    __syncthreads();
    if (wid == 0) {
        float x = (lane < NWAVES) ? shred[lane] : 0.0f;
        x = waveSum(x);
        if (lane == 0) shred[0] = x;
    }
    __syncthreads();
    return shred[0];
}

__global__ __launch_bounds__(BLOCK) void topk_la_row_kernel(
    const float* __restrict__ logit, const int* __restrict__ target,
    const float* __restrict__ lcn, const int* __restrict__ kpc,
    float* __restrict__ row_loss)
{
    __shared__ __align__(16) float sh_logit[NUM_C];
    __shared__ uint32_t sh_hist[256];
    __shared__ float sh_red[NWAVES];
    __shared__ uint32_t sh_prefix;
    __shared__ uint32_t sh_want;

    const int row = blockIdx.x;
    const uint64_t rowbase = (uint64_t)(const void*)(logit + (size_t)row * NUM_C);

    // ---- Stage the row into LDS via CDNA5 async global->LDS DMA (B128) ----
    for (int i = threadIdx.x; i < NUM_C4; i += BLOCK) {
        uint32_t ldsaddr = (uint32_t)(uintptr_t)(&sh_logit[i * 4]);
        uint32_t voff    = (uint32_t)(i * 16);
        asm volatile("global_load_async_to_lds_b128 %0, %1, %2"
                     :: "v"(ldsaddr), "v"(voff), "s"(rowbase) : "memory");
    }
    asm volatile("s_wait_asynccnt 0x0" ::: "memory");
    __syncthreads();

    const float4* sl4  = (const float4*)sh_logit;
    const float4* lcn4 = (const float4*)lcn;

    // ---- Pass A: max of adjusted logits (TAU == 1.0), float4 sweeps ----
    float mloc = -3.402823466e38f;
    for (int i = threadIdx.x; i < NUM_C4; i += BLOCK) {
        float4 a = sl4[i];
        float4 c = lcn4[i];
        mloc = fmaxf(mloc, fmaxf(fmaxf(a.x + c.x, a.y + c.y),
                                 fmaxf(a.z + c.z, a.w + c.w)));
    }
    const float m = blockMax(mloc, sh_red);

    // ---- Pass B: denom = sum exp(adj - m) ----
    float dloc = 0.0f;
    for (int i = threadIdx.x; i < NUM_C4; i += BLOCK) {
        float4 a = sl4[i];
        float4 c = lcn4[i];
        dloc += __expf(a.x + c.x - m);
        dloc += __expf(a.y + c.y - m);
        dloc += __expf(a.z + c.z - m);
        dloc += __expf(a.w + c.w - m);
    }
    const float denom = blockSum(dloc, sh_red);

    // ---- Radix select: k-th largest raw logit (k in [30,100]) ----
    const int t = target[row];
    int k = kpc[t];
    if (k > NUM_C) k = NUM_C;

    uint32_t prefix = 0;
    uint32_t want = (uint32_t)k;   // rank from the top, 1-indexed
#pragma unroll
    for (int shift = 24; shift >= 0; shift -= 8) {
        sh_hist[threadIdx.x] = 0u;
        __syncthreads();
        const uint32_t himask = (shift == 24) ? 0u : (0xFFFFFFFFu << (shift + 8));
        for (int i = threadIdx.x; i < NUM_C4; i += BLOCK) {
            float4 a = sl4[i];
            uint32_t k0 = f2key(a.x), k1 = f2key(a.y), k2 = f2key(a.z), k3 = f2key(a.w);
            if ((k0 & himask) == prefix) atomicAdd(&sh_hist[(k0 >> shift) & 0xFFu], 1u);
            if ((k1 & himask) == prefix) atomicAdd(&sh_hist[(k1 >> shift) & 0xFFu], 1u);
            if ((k2 & himask) == prefix) atomicAdd(&sh_hist[(k2 >> shift) & 0xFFu], 1u);
            if ((k3 & himask) == prefix) atomicAdd(&sh_hist[(k3 >> shift) & 0xFFu], 1u);
        }
        __syncthreads();
        // ---- Parallel bin scan: one wave32, 8 descending bins per lane ----
        if (threadIdx.x < 32) {
            const int lane = threadIdx.x;
            uint32_t h[8];
            uint32_t csum = 0;
#pragma unroll
            for (int j = 0; j < 8; ++j) {
                h[j] = sh_hist[255 - 8 * lane - j];
                csum += h[j];
            }
            // inclusive prefix across lanes (lane order = descending bins)
            uint32_t v = csum;
#pragma unroll
            for (int off = 1; off < 32; off <<= 1) {
                uint32_t y = __shfl_up(v, off, 32);
                if (lane >= off) v += y;
            }
            uint32_t above = v - csum;   // count of elems in bins strictly above this lane's group
            const uint32_t w = want;
#pragma unroll
            for (int j = 0; j < 8; ++j) {
                if (above < w && w <= above + h[j]) {
                    sh_prefix = prefix | ((uint32_t)(255 - 8 * lane - j) << shift);
                    sh_want = w - above;
                }
                above += h[j];
            }
        }
        __syncthreads();
        prefix = sh_prefix;
        want = sh_want;
        __syncthreads();
    }
    const uint32_t thresh_key = prefix;   // key of the k-th largest element

    // ---- Pass C: masked probability mass ----
    float sloc = 0.0f;
    for (int i = threadIdx.x; i < NUM_C4; i += BLOCK) {
        float4 a = sl4[i];
        float4 c = lcn4[i];
        if (f2key(a.x) >= thresh_key) sloc += __expf(a.x + c.x - m);
        if (f2key(a.y) >= thresh_key) sloc += __expf(a.y + c.y - m);
        if (f2key(a.z) >= thresh_key) sloc += __expf(a.z + c.z - m);
        if (f2key(a.w) >= thresh_key) sloc += __expf(a.w + c.w - m);
    }
    const float smasked = blockSum(sloc, sh_red);

    if (threadIdx.x == 0) {
        const float x_t  = sh_logit[t];
        const float adjt = x_t + lcn[t];
        const float logd = __logf(denom);
        const float loss_full = (m + logd) - adjt;               // -log_p_adj[t]
        float p_t = 0.0f;
        if (f2key(x_t) >= thresh_key)
            p_t = __expf(adjt - m) / denom;
        const float S = smasked / denom + 1e-6f * (float)NUM_C;  // sum of masked_prob + eps
        const float loss_topk = __logf(S) - __logf(p_t + 1e-6f);
        row_loss[row] = 0.5f * (loss_full + loss_topk);
    }
}

__global__ __launch_bounds__(256) void topk_la_final_reduce(
    const float* __restrict__ row_loss, float* __restrict__ out)
{
    __shared__ float sh[256];
    float s = 0.0f;
    for (int i = threadIdx.x; i < NUM_B; i += 256)
        s += row_loss[i];
    sh[threadIdx.x] = s;
    __syncthreads();
#pragma unroll
    for (int st = 128; st > 0; st >>= 1) {
        if (threadIdx.x < st) sh[threadIdx.x] += sh[threadIdx.x + st];
        __syncthreads();
    }
    if (threadIdx.x == 0) out[0] = sh[0] / (float)NUM_B;
}

extern "C" void kernel_launch(void* const* d_in, const int* in_sizes, int n_in,
                              void* d_out, int out_size, void* d_ws, size_t ws_size,
                              hipStream_t stream) {
    const float* logit  = (const float*)d_in[0];
    const int*   target = (const int*)d_in[1];
    const float* lcn    = (const float*)d_in[2];
    const int*   kpc    = (const int*)d_in[3];
    float* row_loss = (float*)d_ws;     // 4096 floats of scratch
    float* out      = (float*)d_out;

    topk_la_row_kernel<<<NUM_B, BLOCK, 0, stream>>>(logit, target, lcn, kpc, row_loss);
    topk_la_final_reduce<<<1, 256, 0, stream>>>(row_loss, out);
}